// MambaBlock_72713796321613
// MI455X (gfx1250) — compile-verified
//
#include <hip/hip_runtime.h>
#include <math.h>

typedef __attribute__((ext_vector_type(2))) float v2f;
typedef __attribute__((ext_vector_type(8))) float v8f;

#define L_SEQ   2048
#define DMODEL  1024
#define DINNER  2048
#define DCONV   4
#define DTRANK  64
#define DSTATE  16
#define NPROJ   (DTRANK + 2 * DSTATE)   // 96

__device__ __forceinline__ float silu_f(float x) {
    return x * (1.0f / (1.0f + __expf(-x)));
}
__device__ __forceinline__ float softplus_f(float x) {
    return (x > 20.0f) ? x : log1pf(__expf(x));
}

// ---------------------------------------------------------------------------
// f32 WMMA GEMM:  C[M,N] = act( A[M,K] @ B[K,N] + bias )
// One 32x32 output tile per wave (2x2 grid of 16x16 WMMA fragments),
// K stepped by 4 via V_WMMA_F32_16X16X4_F32. A/B fragments are each reused
// across two WMMAs -> 8 flops/byte of fragment traffic.
// Grid must exactly cover (M/32)*(N/32) tiles with (blockDim.x/32) waves/block.
// act: 0 = none, 1 = softplus
// ---------------------------------------------------------------------------
__global__ void wmma_gemm_f32(const float* __restrict__ A,
                              const float* __restrict__ B,
                              const float* __restrict__ bias,
                              float* __restrict__ C,
                              int N, int K, int lda, int ldb, int ldc, int act)
{
    const int lane    = threadIdx.x & 31;
    const int wave    = threadIdx.x >> 5;
    const int tile    = blockIdx.x * (blockDim.x >> 5) + wave;
    const int tiles_n = N >> 5;
    const int tm      = (tile / tiles_n) << 5;
    const int tn      = (tile % tiles_n) << 5;

    const int half = lane >> 4;        // 0: lanes 0-15, 1: lanes 16-31
    const int sub  = lane & 15;

    // A lane layout (16x4 f32): v0 = A[m][k+2*half], v1 = A[m][k+2*half+1]
    const float* aBase0 = A + (size_t)(tm + sub) * lda + 2 * half;
    const float* aBase1 = aBase0 + (size_t)16 * lda;
    // B lane layout (4x16 f32): v0 = B[k+2*half][n], v1 = B[k+2*half+1][n]
    const float* bBase  = B + (size_t)(2 * half) * ldb + tn + sub;

    v8f c00 = {}, c01 = {}, c10 = {}, c11 = {};
    #pragma unroll 2
    for (int k = 0; k < K; k += 4) {
        v2f a0, a1, b0, b1;
        a0.x = aBase0[k];
        a0.y = aBase0[k + 1];
        a1.x = aBase1[k];
        a1.y = aBase1[k + 1];
        const size_t r0 = (size_t)k * ldb;
        const size_t r1 = (size_t)(k + 1) * ldb;
        b0.x = bBase[r0];
        b0.y = bBase[r1];
        b1.x = bBase[r0 + 16];
        b1.y = bBase[r1 + 16];
        c00 = __builtin_amdgcn_wmma_f32_16x16x4_f32(false, a0, false, b0, (short)0, c00, false, false);
        c01 = __builtin_amdgcn_wmma_f32_16x16x4_f32(false, a0, false, b1, (short)0, c01, false, false);
        c10 = __builtin_amdgcn_wmma_f32_16x16x4_f32(false, a1, false, b0, (short)0, c10, false, false);
        c11 = __builtin_amdgcn_wmma_f32_16x16x4_f32(false, a1, false, b1, (short)0, c11, false, false);
    }

    const int n0 = tn + sub;
    const int n1 = n0 + 16;
    const float bv0 = bias ? bias[n0] : 0.0f;
    const float bv1 = bias ? bias[n1] : 0.0f;
    // D layout: VGPR r -> row (tile_m + r + 8*half), col (tile_n + sub)
    #pragma unroll
    for (int r = 0; r < 8; ++r) {
        const size_t row0 = (size_t)(tm + r + 8 * half) * ldc;
        const size_t row1 = (size_t)(tm + 16 + r + 8 * half) * ldc;
        float v00 = c00[r] + bv0;
        float v01 = c01[r] + bv1;
        float v10 = c10[r] + bv0;
        float v11 = c11[r] + bv1;
        if (act == 1) {
            v00 = softplus_f(v00); v01 = softplus_f(v01);
            v10 = softplus_f(v10); v11 = softplus_f(v11);
        }
        C[row0 + n0] = v00;
        C[row0 + n1] = v01;
        C[row1 + n0] = v10;
        C[row1 + n1] = v11;
    }
}

// ---------------------------------------------------------------------------
// Depthwise causal conv (k=4) over time + SiLU.  xr = [xin | res] (L x 4096).
// One thread per (t, d) element of u (L x 2048). Coalesced across d.
// ---------------------------------------------------------------------------
__global__ void conv_silu_kernel(const float* __restrict__ xr,
                                 const float* __restrict__ conv_w,
                                 const float* __restrict__ conv_b,
                                 float* __restrict__ u)
{
    const int idx = blockIdx.x * blockDim.x + threadIdx.x;  // over L*DINNER
    const int d   = idx & (DINNER - 1);
    const int t   = idx >> 11;                              // DINNER == 2^11
    float acc = conv_b[d];
    #pragma unroll
    for (int j = 0; j < DCONV; ++j) {
        const int tt = t - (DCONV - 1) + j;
        const float xv = (tt >= 0) ? xr[(size_t)tt * (2 * DINNER) + d] : 0.0f;
        acc = fmaf(xv, conv_w[j * DINNER + d], acc);
    }
    u[idx] = silu_f(acc);
}

// ---------------------------------------------------------------------------
// Selective scan: one thread per channel d, 16 states in registers,
// sequential over t. B(t)/C(t) (broadcast across channels) staged in LDS in
// 128-step chunks. Fused epilogue: y = (scan + u*D) * silu(res).
// ---------------------------------------------------------------------------
#define TCHUNK 128
__global__ void scan_kernel(const float* __restrict__ delta,
                            const float* __restrict__ u,
                            const float* __restrict__ dbc,
                            const float* __restrict__ A_log,
                            const float* __restrict__ Dp,
                            const float* __restrict__ xr,
                            float* __restrict__ y)
{
    __shared__ float sBC[TCHUNK][2 * DSTATE];   // 16 KB
    const int d = blockIdx.x * blockDim.x + threadIdx.x;   // 0..2047

    float Arow[DSTATE];
    #pragma unroll
    for (int n = 0; n < DSTATE; ++n)
        Arow[n] = -__expf(A_log[d * DSTATE + n]);
    const float Dd = Dp[d];

    float h[DSTATE];
    #pragma unroll
    for (int n = 0; n < DSTATE; ++n) h[n] = 0.0f;

    for (int t0 = 0; t0 < L_SEQ; t0 += TCHUNK) {
        __syncthreads();   // protect sBC from previous chunk's readers
        for (int i = threadIdx.x; i < TCHUNK * 2 * DSTATE; i += blockDim.x) {
            const int tt = i >> 5;
            const int c  = i & 31;
            sBC[tt][c] = dbc[(size_t)(t0 + tt) * NPROJ + DTRANK + c];
        }
        __syncthreads();

        for (int ti = 0; ti < TCHUNK; ++ti) {
            const int t    = t0 + ti;
            const float dl = delta[(size_t)t * DINNER + d];
            const float ut = u[(size_t)t * DINNER + d];
            const float du = dl * ut;
            float yt = 0.0f;
            #pragma unroll
            for (int n = 0; n < DSTATE; ++n) {
                const float dA = __expf(dl * Arow[n]);
                h[n] = fmaf(dA, h[n], du * sBC[ti][n]);
                yt   = fmaf(h[n], sBC[ti][DSTATE + n], yt);
            }
            const float r = xr[(size_t)t * (2 * DINNER) + DINNER + d];
            y[(size_t)t * DINNER + d] = (yt + ut * Dd) * silu_f(r);
        }
    }
}

// ---------------------------------------------------------------------------
extern "C" void kernel_launch(void* const* d_in, const int* in_sizes, int n_in,
                              void* d_out, int out_size, void* d_ws, size_t ws_size,
                              hipStream_t stream)
{
    const float* x          = (const float*)d_in[0];
    const float* in_proj_w  = (const float*)d_in[1];
    const float* in_proj_b  = (const float*)d_in[2];
    const float* conv_w     = (const float*)d_in[3];
    const float* conv_b     = (const float*)d_in[4];
    const float* x_proj_w   = (const float*)d_in[5];
    const float* dt_proj_w  = (const float*)d_in[6];
    const float* dt_proj_b  = (const float*)d_in[7];
    const float* A_log      = (const float*)d_in[8];
    const float* Dp         = (const float*)d_in[9];
    const float* out_proj_w = (const float*)d_in[10];
    const float* out_proj_b = (const float*)d_in[11];
    float* out              = (float*)d_out;

    // Workspace layout (floats)
    float* ws    = (float*)d_ws;
    float* xr    = ws;                                   // L x 4096  (32 MB)
    float* u     = xr    + (size_t)L_SEQ * 2 * DINNER;   // L x 2048  (16 MB)
    float* dbc   = u     + (size_t)L_SEQ * DINNER;       // L x 96
    float* delta = dbc   + (size_t)L_SEQ * NPROJ;        // L x 2048  (16 MB)
    float* y     = delta + (size_t)L_SEQ * DINNER;       // L x 2048  (16 MB)

    const int BLK = 256;            // 8 waves/block
    const int WPB = BLK / 32;

    // 1) xr = x @ in_proj_w + b          (2048x1024 @ 1024x4096)
    {
        int tiles = (L_SEQ / 32) * (2 * DINNER / 32);    // 8192
        wmma_gemm_f32<<<tiles / WPB, BLK, 0, stream>>>(
            x, in_proj_w, in_proj_b, xr,
            2 * DINNER, DMODEL, DMODEL, 2 * DINNER, 2 * DINNER, 0);
    }

    // 2) u = silu(depthwise_conv(xin) + conv_b)
    conv_silu_kernel<<<(L_SEQ * DINNER) / BLK, BLK, 0, stream>>>(xr, conv_w, conv_b, u);

    // 3) dbc = u @ x_proj_w              (2048x2048 @ 2048x96)
    {
        int tiles = (L_SEQ / 32) * (NPROJ / 32);         // 192
        wmma_gemm_f32<<<tiles / WPB, BLK, 0, stream>>>(
            u, x_proj_w, nullptr, dbc,
            NPROJ, DINNER, DINNER, NPROJ, NPROJ, 0);
    }

    // 4) delta = softplus(dt @ dt_proj_w + dt_proj_b)   (2048x64 @ 64x2048)
    {
        int tiles = (L_SEQ / 32) * (DINNER / 32);        // 4096
        wmma_gemm_f32<<<tiles / WPB, BLK, 0, stream>>>(
            dbc, dt_proj_w, dt_proj_b, delta,
            DINNER, DTRANK, NPROJ, DINNER, DINNER, 1);
    }

    // 5) selective scan + fused (u*D, silu(res)) epilogue
    scan_kernel<<<DINNER / BLK, BLK, 0, stream>>>(delta, u, dbc, A_log, Dp, xr, y);

    // 6) out = y @ out_proj_w + out_proj_b   (2048x2048 @ 2048x1024)
    {
        int tiles = (L_SEQ / 32) * (DMODEL / 32);        // 2048
        wmma_gemm_f32<<<tiles / WPB, BLK, 0, stream>>>(
            y, out_proj_w, out_proj_b, out,
            DMODEL, DINNER, DINNER, DMODEL, DMODEL, 0);
    }
}